// RnnAutoEncoder_44650480009505
// MI455X (gfx1250) — compile-verified
//
#include <hip/hip_runtime.h>
#include <math.h>

typedef float v2f __attribute__((ext_vector_type(2)));
typedef float v8f __attribute__((ext_vector_type(8)));
typedef int   v4i __attribute__((vector_size(16)));     // matches builtin's pointee type

// ---- feature guards (evaluated per compilation pass) ----
#if defined(__has_builtin)
#if __has_builtin(__builtin_amdgcn_global_load_async_to_lds_b128) && \
    __has_builtin(__builtin_amdgcn_s_wait_asynccnt)
#define HAVE_ASYNC_LDS 1
#endif
#if __has_builtin(__builtin_amdgcn_tanhf)
#define fast_tanh __builtin_amdgcn_tanhf
#endif
#endif
#ifndef fast_tanh
#define fast_tanh tanhf
#endif

#define T_  512
#define B_  256
#define IN_ 64
#define M_  (T_ * B_)          // 131072 rows total

// ---- workspace layout (float offsets) ----
// Weight buffers are stored PAIR-INTERLEAVED for WMMA B-fragments:
//   WP[((k>>1)*N + n)*2 + (k&1)] = B[k][n]   (B = W^T, K x N, zero padded)
// so a fragment {B[k][n], B[k+1][n]} is one aligned 8-byte load.
#define OFF_P      0                         // (T*B) x 32 : xp1, then overwritten in place with relu(h2)
#define NP_        (T_ * B_ * 32)
#define OFF_WE1    (NP_)                     // 64x128  = e_w1^T
#define OFF_WE2    (OFF_WE1 + 64*128)        // 128x64  = e_w2^T
#define OFF_WE3    (OFF_WE2 + 128*64)        // 64x32   = e_w3^T
#define OFF_WR1IH  (OFF_WE3 + 64*32)         // 32x32   = r1_wih^T (N padded 20->32)
#define OFF_WR1HH  (OFF_WR1IH + 32*32)       // 32x32   = r1_whh^T (K,N padded 20->32)
#define OFF_WR2IH  (OFF_WR1HH + 32*32)       // 32x32   = r2_wih^T (K padded 20->32)
#define OFF_WR2HH  (OFF_WR2IH + 32*32)       // 32x32   = r2_whh^T
#define OFF_WD1    (OFF_WR2HH + 32*32)       // 32x64   = d_w1^T
#define OFF_WD2    (OFF_WD1 + 32*64)         // 64x128  = d_w2^T
#define OFF_WD3    (OFF_WD2 + 64*128)        // 128x64  = d_w3^T
#define OFF_BE1    (OFF_WD3 + 128*64)        // 128
#define OFF_BE2    (OFF_BE1 + 128)           // 64
#define OFF_BE3    (OFF_BE2 + 64)            // 32
#define OFF_BR1    (OFF_BE3 + 32)            // 32 : r1_bih + r1_bhh (padded)
#define OFF_BR2    (OFF_BR1 + 32)            // 32 : r2_bih + r2_bhh
#define OFF_BD1    (OFF_BR2 + 32)            // 64
#define OFF_BD2    (OFF_BD1 + 64)            // 128
#define OFF_BD3    (OFF_BD2 + 128)           // 64

// interleaved index for B[k][n] in a K x N transposed-weight buffer
__device__ __forceinline__ int ilv(int k, int n, int N) {
    return ((k >> 1) * N + n) * 2 + (k & 1);
}

// ---------------------------------------------------------------------------
// Prep: build pair-interleaved transposed weights + fused biases in ws.
// ---------------------------------------------------------------------------
__global__ void prep_kernel(const float* e_w1, const float* e_b1,
                            const float* e_w2, const float* e_b2,
                            const float* e_w3, const float* e_b3,
                            const float* r1_wih, const float* r1_whh,
                            const float* r1_bih, const float* r1_bhh,
                            const float* r2_wih, const float* r2_whh,
                            const float* r2_bih, const float* r2_bhh,
                            const float* d_w1, const float* d_b1,
                            const float* d_w2, const float* d_b2,
                            const float* d_w3, const float* d_b3,
                            float* ws)
{
    int tid = blockIdx.x * blockDim.x + threadIdx.x;
    int nth = gridDim.x * blockDim.x;
    for (int i = tid; i < 64*128; i += nth) { int k=i/128,n=i%128; ws[OFF_WE1+ilv(k,n,128)] = e_w1[n*64+k]; }
    for (int i = tid; i < 128*64; i += nth) { int k=i/64, n=i%64;  ws[OFF_WE2+ilv(k,n,64)]  = e_w2[n*128+k]; }
    for (int i = tid; i < 64*32;  i += nth) { int k=i/32, n=i%32;  ws[OFF_WE3+ilv(k,n,32)]  = e_w3[n*64+k]; }
    for (int i = tid; i < 32*32;  i += nth) { int k=i/32, n=i%32;  ws[OFF_WR1IH+ilv(k,n,32)] = (n<20) ? r1_wih[n*32+k] : 0.f; }
    for (int i = tid; i < 32*32;  i += nth) { int k=i/32, n=i%32;  ws[OFF_WR1HH+ilv(k,n,32)] = (k<20 && n<20) ? r1_whh[n*20+k] : 0.f; }
    for (int i = tid; i < 32*32;  i += nth) { int k=i/32, n=i%32;  ws[OFF_WR2IH+ilv(k,n,32)] = (k<20) ? r2_wih[n*20+k] : 0.f; }
    for (int i = tid; i < 32*32;  i += nth) { int k=i/32, n=i%32;  ws[OFF_WR2HH+ilv(k,n,32)] = r2_whh[n*32+k]; }
    for (int i = tid; i < 32*64;  i += nth) { int k=i/64, n=i%64;  ws[OFF_WD1+ilv(k,n,64)]  = d_w1[n*32+k]; }
    for (int i = tid; i < 64*128; i += nth) { int k=i/128,n=i%128; ws[OFF_WD2+ilv(k,n,128)] = d_w2[n*64+k]; }
    for (int i = tid; i < 128*64; i += nth) { int k=i/64, n=i%64;  ws[OFF_WD3+ilv(k,n,64)]  = d_w3[n*128+k]; }
    for (int i = tid; i < 128; i += nth) ws[OFF_BE1+i] = e_b1[i];
    for (int i = tid; i < 64;  i += nth) ws[OFF_BE2+i] = e_b2[i];
    for (int i = tid; i < 32;  i += nth) ws[OFF_BE3+i] = e_b3[i];
    for (int i = tid; i < 32;  i += nth) ws[OFF_BR1+i] = (i<20) ? (r1_bih[i] + r1_bhh[i]) : 0.f;
    for (int i = tid; i < 32;  i += nth) ws[OFF_BR2+i] = r2_bih[i] + r2_bhh[i];
    for (int i = tid; i < 64;  i += nth) ws[OFF_BD1+i] = d_b1[i];
    for (int i = tid; i < 128; i += nth) ws[OFF_BD2+i] = d_b2[i];
    for (int i = tid; i < 64;  i += nth) ws[OFF_BD3+i] = d_b3[i];
}

// ---------------------------------------------------------------------------
// Stage a tile of `nf4` float4s from global into LDS (one wave).
// Uses CDNA5 async global->LDS DMA (ASYNCcnt) when the toolchain exposes it.
// ---------------------------------------------------------------------------
__device__ __attribute__((always_inline))
void stage_tile(const float* __restrict__ g, float* s, int nf4, unsigned lane)
{
#if defined(HAVE_ASYNC_LDS)
    typedef __attribute__((address_space(1))) v4i gv4i_t;   // global (__device__) int4
    typedef __attribute__((address_space(3))) v4i lv4i_t;   // LDS (__shared__) int4
    for (int i = lane; i < nf4; i += 32) {
        gv4i_t* gp = (gv4i_t*)(unsigned long long)(const void*)(g + i*4);
        lv4i_t* lp = (lv4i_t*)(unsigned)(unsigned long long)(void*)(s + i*4);
        __builtin_amdgcn_global_load_async_to_lds_b128(gp, lp, 0, 0);
    }
    __builtin_amdgcn_s_wait_asynccnt(0);
#else
    const float4* gs = (const float4*)g;
    float4* sd = (float4*)s;
    for (int i = lane; i < nf4; i += 32) sd[i] = gs[i];
#endif
}

// ---------------------------------------------------------------------------
// One 16-row GEMM tile with V_WMMA_F32_16X16X4_F32.
// A: LDS, 16 x K row-major.  B: global, pair-interleaved K x N (transposed).
// Out: LDS, 16 x N row-major.  Bias broadcast per column, optional ReLU.
//
// f32 16x16x4 fragment layout (ISA 7.12.2):
//   A: lane<16 -> {A[m][4c+0], A[m][4c+1]}, lane>=16 -> {A[m][4c+2], A[m][4c+3]}, m=lane&15
//   B: lane<16 -> {B[4c+0][n], B[4c+1][n]}, lane>=16 -> {B[4c+2][n], B[4c+3][n]}, n=lane&15
//   C/D: vgpr r -> M = r + 8*(lane>>4), N = lane&15
// ---------------------------------------------------------------------------
template<int K, int N, bool RELU>
__device__ __attribute__((always_inline))
void gemm_tile(const float* S, const float* WT, const float* bias, float* Sout,
               unsigned lane)
{
    const unsigned half = lane >> 4;
    const unsigned l16  = lane & 15u;
#pragma unroll
    for (int j = 0; j < N/16; ++j) {
        v8f c;
        float bz = bias[j*16 + l16];
#pragma unroll
        for (int r = 0; r < 8; ++r) c[r] = bz;
#pragma unroll
        for (int kc = 0; kc < K/4; ++kc) {
            v2f a, b;
            a.x = S[l16*K + 4*kc + 2*half];
            a.y = S[l16*K + 4*kc + 2*half + 1];
            // pair p = 2*kc + half holds rows {4kc+2h, 4kc+2h+1}: one b64 load
            b = *(const v2f*)(WT + 2 * ((2*kc + half)*N + j*16 + l16));
            c = __builtin_amdgcn_wmma_f32_16x16x4_f32(
                    false, a, false, b, (short)0, c, false, false);
        }
#pragma unroll
        for (int r = 0; r < 8; ++r) {
            float v = c[r];
            if (RELU) v = fmaxf(v, 0.0f);
            Sout[(r + 8*half)*N + j*16 + l16] = v;
        }
    }
}

// ---------------------------------------------------------------------------
// Encoder MLP + fused RNN1 input projection. One wave per 16-row tile.
// x(16x64) -> relu 128 -> relu 64 -> relu 32 -> xp1(16x32, cols>=20 are 0) -> P
// ---------------------------------------------------------------------------
__global__ __launch_bounds__(32) void encoder_kernel(const float* __restrict__ x,
                                                     float* __restrict__ ws)
{
    __shared__ __align__(16) float S0[16*128];
    __shared__ __align__(16) float S1[16*128];
    const unsigned lane = threadIdx.x;
    const size_t tile = blockIdx.x;

    stage_tile(x + tile * (16*64), S0, 16*64/4, lane);
    __syncthreads();

    gemm_tile< 64, 128, true>(S0, ws + OFF_WE1, ws + OFF_BE1, S1, lane);
    __syncthreads();
    gemm_tile<128,  64, true>(S1, ws + OFF_WE2, ws + OFF_BE2, S0, lane);
    __syncthreads();
    gemm_tile< 64,  32, true>(S0, ws + OFF_WE3, ws + OFF_BE3, S1, lane);
    __syncthreads();
    // xp1 = h3 @ r1_wih^T + (bih+bhh); padded columns come out exactly 0
    gemm_tile< 32,  32, false>(S1, ws + OFF_WR1IH, ws + OFF_BR1, S0, lane);
    __syncthreads();

    float4* pt = (float4*)(ws + OFF_P + tile * (16*32));
    const float4* ss = (const float4*)S0;
    for (int i = lane; i < 16*32/4; i += 32) pt[i] = ss[i];
}

// ---------------------------------------------------------------------------
// Fused double RNN scan. 16 one-wave blocks; block b owns batch rows 16b..16b+15
// for all 512 timesteps (batch-parallel, zero inter-wave communication).
// Recurrent B-fragments preloaded into VGPRs; per-step LDS bounce converts
// C-layout results back into A-fragments. Overwrites P in place with relu(h2).
// ---------------------------------------------------------------------------
__global__ __launch_bounds__(32) void rnn_kernel(float* __restrict__ ws)
{
    __shared__ __align__(16) float R[16*32];   // wave-private re-fragment buffer
    const unsigned lane = threadIdx.x;
    const unsigned half = lane >> 4;
    const unsigned l16  = lane & 15u;
    const int rowbase = blockIdx.x * 16;

    // ---- preload time-invariant B fragments (pair-interleaved: one b64 each) ----
    v2f Br1hh[5][2], Br2ih[5][2], Br2hh[8][2];
    float bz2[2];
#pragma unroll
    for (int j = 0; j < 2; ++j) {
#pragma unroll
        for (int kc = 0; kc < 5; ++kc) {
            Br1hh[kc][j] = *(const v2f*)(ws + OFF_WR1HH + 2*((2*kc + half)*32 + j*16 + l16));
            Br2ih[kc][j] = *(const v2f*)(ws + OFF_WR2IH + 2*((2*kc + half)*32 + j*16 + l16));
        }
#pragma unroll
        for (int kc = 0; kc < 8; ++kc)
            Br2hh[kc][j] = *(const v2f*)(ws + OFF_WR2HH + 2*((2*kc + half)*32 + j*16 + l16));
        bz2[j] = ws[OFF_BR2 + j*16 + l16];
    }

    v2f a1[5], a2[8];                     // A-fragments of h1 (K=20) and h2 (K=32)
#pragma unroll
    for (int c = 0; c < 5; ++c) a1[c] = (v2f)0.f;   // h0 = 0
#pragma unroll
    for (int c = 0; c < 8; ++c) a2[c] = (v2f)0.f;

    for (int t = 0; t < T_; ++t) {
        float* pt = ws + OFF_P + ((size_t)t * B_ + rowbase) * 32;
        if (t + 1 < T_)
            __builtin_prefetch(ws + OFF_P + ((size_t)(t+1) * B_ + rowbase) * 32, 0, 1);

        // ---- RNN1: h1 = tanh(xp1[t] + h1_prev @ Whh1^T) ----
        v8f c1[2];
#pragma unroll
        for (int j = 0; j < 2; ++j) {
#pragma unroll
            for (int r = 0; r < 8; ++r)
                c1[j][r] = pt[(r + 8*half)*32 + j*16 + l16];   // xp includes bias
#pragma unroll
            for (int kc = 0; kc < 5; ++kc)
                c1[j] = __builtin_amdgcn_wmma_f32_16x16x4_f32(
                            false, a1[kc], false, Br1hh[kc][j], (short)0, c1[j], false, false);
#pragma unroll
            for (int r = 0; r < 8; ++r) c1[j][r] = fast_tanh(c1[j][r]);
        }
        // C-layout -> LDS -> A-fragments of new h1 (cols >=20 are tanh(0)=0, never read)
#pragma unroll
        for (int j = 0; j < 2; ++j)
#pragma unroll
            for (int r = 0; r < 8; ++r)
                R[(r + 8*half)*32 + j*16 + l16] = c1[j][r];
#pragma unroll
        for (int kc = 0; kc < 5; ++kc) {
            a1[kc].x = R[l16*32 + 4*kc + 2*half];
            a1[kc].y = R[l16*32 + 4*kc + 2*half + 1];
        }

        // ---- RNN2: h2 = tanh(h1 @ Wih2^T + bias2 + h2_prev @ Whh2^T) ----
        v8f c2[2];
#pragma unroll
        for (int j = 0; j < 2; ++j) {
#pragma unroll
            for (int r = 0; r < 8; ++r) c2[j][r] = bz2[j];
#pragma unroll
            for (int kc = 0; kc < 5; ++kc)
                c2[j] = __builtin_amdgcn_wmma_f32_16x16x4_f32(
                            false, a1[kc], false, Br2ih[kc][j], (short)0, c2[j], false, false);
#pragma unroll
            for (int kc = 0; kc < 8; ++kc)
                c2[j] = __builtin_amdgcn_wmma_f32_16x16x4_f32(
                            false, a2[kc], false, Br2hh[kc][j], (short)0, c2[j], false, false);
#pragma unroll
            for (int r = 0; r < 8; ++r) c2[j][r] = fast_tanh(c2[j][r]);
        }
        // decoder input: relu(h2) -> P[t] (in place over consumed xp1[t])
#pragma unroll
        for (int j = 0; j < 2; ++j)
#pragma unroll
            for (int r = 0; r < 8; ++r)
                pt[(r + 8*half)*32 + j*16 + l16] = fmaxf(c2[j][r], 0.0f);
        // h2 -> A-fragments for next step
#pragma unroll
        for (int j = 0; j < 2; ++j)
#pragma unroll
            for (int r = 0; r < 8; ++r)
                R[(r + 8*half)*32 + j*16 + l16] = c2[j][r];
#pragma unroll
        for (int kc = 0; kc < 8; ++kc) {
            a2[kc].x = R[l16*32 + 4*kc + 2*half];
            a2[kc].y = R[l16*32 + 4*kc + 2*half + 1];
        }
    }
}

// ---------------------------------------------------------------------------
// Decoder MLP: relu(h2)(16x32) -> relu 64 -> relu 128 -> 64 (+bias) -> d_out
// ---------------------------------------------------------------------------
__global__ __launch_bounds__(32) void decoder_kernel(const float* __restrict__ ws,
                                                     float* __restrict__ out)
{
    __shared__ __align__(16) float S0[16*128];
    __shared__ __align__(16) float S1[16*128];
    const unsigned lane = threadIdx.x;
    const size_t tile = blockIdx.x;

    stage_tile(ws + OFF_P + tile * (16*32), S0, 16*32/4, lane);
    __syncthreads();

    gemm_tile< 32,  64, true >(S0, ws + OFF_WD1, ws + OFF_BD1, S1, lane);
    __syncthreads();
    gemm_tile< 64, 128, true >(S1, ws + OFF_WD2, ws + OFF_BD2, S0, lane);
    __syncthreads();
    gemm_tile<128,  64, false>(S0, ws + OFF_WD3, ws + OFF_BD3, S1, lane);
    __syncthreads();

    float4* ot = (float4*)(out + tile * (16*64));
    const float4* ss = (const float4*)S1;
    for (int i = lane; i < 16*64/4; i += 32) ot[i] = ss[i];
}

// ---------------------------------------------------------------------------
extern "C" void kernel_launch(void* const* d_in, const int* in_sizes, int n_in,
                              void* d_out, int out_size, void* d_ws, size_t ws_size,
                              hipStream_t stream) {
    (void)in_sizes; (void)n_in; (void)out_size; (void)ws_size;
    const float* x      = (const float*)d_in[0];
    const float* e_w1   = (const float*)d_in[1];
    const float* e_b1   = (const float*)d_in[2];
    const float* e_w2   = (const float*)d_in[3];
    const float* e_b2   = (const float*)d_in[4];
    const float* e_w3   = (const float*)d_in[5];
    const float* e_b3   = (const float*)d_in[6];
    const float* r1_wih = (const float*)d_in[7];
    const float* r1_whh = (const float*)d_in[8];
    const float* r1_bih = (const float*)d_in[9];
    const float* r1_bhh = (const float*)d_in[10];
    const float* r2_wih = (const float*)d_in[11];
    const float* r2_whh = (const float*)d_in[12];
    const float* r2_bih = (const float*)d_in[13];
    const float* r2_bhh = (const float*)d_in[14];
    const float* d_w1   = (const float*)d_in[15];
    const float* d_b1   = (const float*)d_in[16];
    const float* d_w2   = (const float*)d_in[17];
    const float* d_b2   = (const float*)d_in[18];
    const float* d_w3   = (const float*)d_in[19];
    const float* d_b3   = (const float*)d_in[20];
    float* ws   = (float*)d_ws;
    float* outp = (float*)d_out;

    prep_kernel<<<64, 256, 0, stream>>>(e_w1, e_b1, e_w2, e_b2, e_w3, e_b3,
                                        r1_wih, r1_whh, r1_bih, r1_bhh,
                                        r2_wih, r2_whh, r2_bih, r2_bhh,
                                        d_w1, d_b1, d_w2, d_b2, d_w3, d_b3, ws);
    encoder_kernel<<<M_/16, 32, 0, stream>>>(x, ws);
    rnn_kernel<<<B_/16, 32, 0, stream>>>(ws);
    decoder_kernel<<<M_/16, 32, 0, stream>>>(ws, outp);
}